// BasicMFNet_6665789243440
// MI455X (gfx1250) — compile-verified
//
#include <hip/hip_runtime.h>
#include <hip/hip_bf16.h>

typedef __attribute__((ext_vector_type(2))) float v2f;
typedef __attribute__((ext_vector_type(4))) float v4f;
typedef __attribute__((ext_vector_type(8))) float v8f;

#define HIDDEN 64
#define NT 4            // N-tiles per wave (A-fragment reuse factor)

// ---------------------------------------------------------------------------
// Kernel 1: zero the label region (nontemporal b128), zero masks, write ratio
// ---------------------------------------------------------------------------
__global__ __launch_bounds__(256) void mf_zero(float* __restrict__ label,
                                               size_t n4,
                                               float* __restrict__ masks,
                                               int nmask,
                                               float* __restrict__ ratio_out,
                                               float ratio) {
    size_t i = (size_t)blockIdx.x * blockDim.x + threadIdx.x;
    size_t stride = (size_t)gridDim.x * blockDim.x;
    v4f z = {0.0f, 0.0f, 0.0f, 0.0f};
    v4f* l4 = (v4f*)label;
    for (size_t j = i; j < n4; j += stride) {
        __builtin_nontemporal_store(z, l4 + j);
    }
    if (i < (size_t)nmask) masks[i] = 0.0f;
    if (i == 0) *ratio_out = ratio;
}

// ---------------------------------------------------------------------------
// Kernel 2: scatter — build row masks and scatter-add ratings into label
// ---------------------------------------------------------------------------
__global__ __launch_bounds__(256) void mf_scatter(const int* __restrict__ idx,
                                                  const float* __restrict__ ratings,
                                                  float* __restrict__ um,
                                                  float* __restrict__ im,
                                                  float* __restrict__ label,
                                                  int nnz, int Ni) {
    int n = blockIdx.x * blockDim.x + threadIdx.x;
    if (n >= nnz) return;
    int u = idx[n];
    int i = idx[nnz + n];
    um[u] = 1.0f;                       // benign write races: all write 1.0
    im[i] = 1.0f;
    atomicAdd(label + (size_t)u * Ni + i, ratings[n]);  // duplicates sum
}

// ---------------------------------------------------------------------------
// Kernel 3: pred = (um .* eu) @ (im .* ei)^T via V_WMMA_F32_16X16X4_F32
//
// Each wave owns a 16 x (16*NT) output strip: the 16 A fragments (K=64,
// 2 VGPRs each, pre-scaled by the user mask) stay resident in registers and
// are reused across NT consecutive N-tiles, cutting L1/L2 read traffic from
// 8 B/loaded per B stored to ~5 B/B.
//
// f32 16x4 A-fragment layout (ISA 7.12.2): lane holds row M = lane&15,
// VGPR pair = K elements {k0+2*(lane>>4), k0+2*(lane>>4)+1} -> contiguous
// float2 load from the embedding row. B (4x16) is symmetric with rows of
// ei (since B[k][n] = ei[n][k]), so it is also a contiguous float2 load.
// C/D layout: lane L, vgpr v -> element (M = v + 8*(L>>4), N = L&15).
// ---------------------------------------------------------------------------
__global__ __launch_bounds__(256) void mf_gemm(const float* __restrict__ eu,
                                               const float* __restrict__ ei,
                                               const float* __restrict__ um,
                                               const float* __restrict__ im,
                                               float* __restrict__ pred,
                                               int Ni) {
    const int lane = threadIdx.x & 31;
    const int wave = threadIdx.x >> 5;
    const int lm = lane & 15;
    const int hi = lane >> 4;

    const int tile_n0 = (blockIdx.x * 8 + wave) * NT;  // first N-tile of strip
    const int m_base  = blockIdx.y * 16;

    // ---- load + mask-scale all 16 A fragments once (32 VGPRs resident) ----
    const float* arow = eu + (size_t)(m_base + lm) * HIDDEN + 2 * hi;
    const float umask = um[m_base + lm];
    v2f afrag[16];
#pragma unroll
    for (int kk = 0; kk < 16; ++kk) {
        v2f a = *(const v2f*)(arow + kk * 4);
        a.x *= umask;
        a.y *= umask;
        afrag[kk] = a;
    }

    // ---- sweep NT N-tiles reusing the A fragments ----
#pragma unroll
    for (int t = 0; t < NT; ++t) {
        const int n_base = (tile_n0 + t) * 16;
        const float* brow = ei + (size_t)(n_base + lm) * HIDDEN + 2 * hi;
        const float imask = im[n_base + lm];

        v8f c = {};
#pragma unroll
        for (int kk = 0; kk < 16; ++kk) {      // k0 = 4*kk
            v2f b = *(const v2f*)(brow + kk * 4);
            c = __builtin_amdgcn_wmma_f32_16x16x4_f32(
                /*neg_a=*/false, afrag[kk], /*neg_b=*/false, b,
                /*c_mod=*/(short)0, c, /*reuse_a=*/false, /*reuse_b=*/false);
        }

        float* outp = pred + (size_t)m_base * Ni + n_base + lm;
#pragma unroll
        for (int v = 0; v < 8; ++v) {
            // lanes 0-15 write row M=v, lanes 16-31 row M=v+8 (64B segments)
            __builtin_nontemporal_store(c[v] * imask,
                                        outp + (size_t)(v + 8 * hi) * Ni);
        }
    }
}

// ---------------------------------------------------------------------------
extern "C" void kernel_launch(void* const* d_in, const int* in_sizes, int n_in,
                              void* d_out, int out_size, void* d_ws, size_t ws_size,
                              hipStream_t stream) {
    const float* eu      = (const float*)d_in[0];   // [U, 64] f32
    const float* ei      = (const float*)d_in[1];   // [I, 64] f32
    const int*   idx     = (const int*)d_in[2];     // [2, NNZ] int
    const float* ratings = (const float*)d_in[3];   // [NNZ] f32

    const int U   = in_sizes[0] / HIDDEN;           // 8192
    const int I   = in_sizes[1] / HIDDEN;           // 16384
    const int NNZ = in_sizes[3];                    // 200000

    float* pred      = (float*)d_out;               // [U, I]
    float* label     = pred + (size_t)U * I;        // [U, I]
    float* ratio_out = label + (size_t)U * I;       // [1]

    float* um = (float*)d_ws;                       // [U] user row mask
    float* im = um + U;                             // [I] item row mask

    const float ratio = (float)((double)U * (double)I / (double)NNZ);

    // 1) zero label + masks, write ratio
    mf_zero<<<2048, 256, 0, stream>>>(label, (size_t)U * I / 4, um, U + I,
                                      ratio_out, ratio);
    // 2) masks + label scatter-add
    mf_scatter<<<(NNZ + 255) / 256, 256, 0, stream>>>(idx, ratings, um, im,
                                                      label, NNZ, I);
    // 3) masked GEMM: each wave = 16x(16*NT) strip, 8 waves/block
    //    grid = (N/16/NT/8, M/16) = (32, 512)
    mf_gemm<<<dim3(I / 16 / NT / 8, U / 16), 256, 0, stream>>>(eu, ei, um, im,
                                                               pred, I);
}